// ArcFaceLoss_3204045603870
// MI455X (gfx1250) — compile-verified
//
#include <hip/hip_runtime.h>
#include <stdint.h>

// ---------------- ArcFace constants (match reference) ----------------
#define SCALING 32.0f
#define COS_M_C 0.8775825618903728f   // cos(0.5)
#define SIN_M_C 0.4794255386042030f   // sin(0.5)
#define TH_C   (-0.8775825618903728f) // cos(pi - 0.5)
#define MM_C    0.2397127693021015f   // sin(pi - 0.5) * 0.5

// ---------------- pipeline geometry ----------------
#define BLK     256                  // 8 wave32s
#define K_LD    4                    // async b64 loads per thread per tile
#define TILE_F2 (BLK * K_LD)         // 1024 float2 = 2048 floats = 8 KB per tile
#define PIPE    4                    // LDS ring depth (32 KB)

// Wait until at most `lit` async ops remain outstanding for this wave.
#define S_WAIT_ASYNC(lit) asm volatile("s_wait_asynccnt " #lit ::: "memory")
// Drain this wave's DS reads before the async engine may overwrite the buffer.
#define S_WAIT_DS0()      asm volatile("s_wait_dscnt 0" ::: "memory")

// gfx1250 async global->LDS copy, GVS mode: saddr(64b SGPR base) + vgpr byte offset.
// VDST vgpr = LDS byte address, VADDR vgpr = byte offset, SADDR = base.
__device__ __forceinline__ void async_ld_f2(uint32_t lds_byte, uint32_t gbyte,
                                            const float* base) {
  asm volatile("global_load_async_to_lds_b64 %0, %1, %2"
               :: "v"(lds_byte), "v"(gbyte), "s"(base)
               : "memory");
}

// ArcFace per-element contribution: exp(logit - 32), with margin at target col.
__device__ __forceinline__ float elem_term(float v, int j, int target,
                                           float* s_tgt) {
  float c  = fminf(fmaxf(v, -1.0f), 1.0f);
  float lg = SCALING * c;
  if (j == target) {
    float sq = fminf(fmaxf(1.0f - c * c, 0.0f), 1.0f);
    float sn = sqrtf(sq);
    float ph = c * COS_M_C - sn * SIN_M_C;
    ph = (c > TH_C) ? ph : (c - MM_C);
    lg = SCALING * ph;
    *s_tgt = lg;                 // exactly one thread per block writes this
  }
  return __expf(lg - SCALING);   // lg <= 32 always -> no overflow; >= exp(-64)
}

__global__ __launch_bounds__(BLK) void arcface_row_kernel(
    const float* __restrict__ cosine,
    const int*   __restrict__ targets,
    float*       __restrict__ row_loss,
    int ncols)
{
  __shared__ __align__(16) float lbuf[PIPE * TILE_F2 * 2];  // 32 KB ring
  __shared__ float red[BLK];
  __shared__ float s_tgt;

  const int   row    = blockIdx.x;
  const int   tid    = threadIdx.x;
  const float* base  = cosine + (size_t)row * (size_t)ncols;
  const int   target = targets[row];
  const int   nf2    = ncols >> 1;
  const int   ntile  = (nf2 + TILE_F2 - 1) / TILE_F2;

  // On AMDGPU the low 32 bits of a generic (flat) pointer into LDS are the
  // wave-relative LDS byte address (ISA 10.2 aperture mapping).
  const uint32_t lds_base = (uint32_t)(uintptr_t)(void*)lbuf;
  const float2*  lbuf2    = reinterpret_cast<const float2*>(lbuf);

  float acc = 0.0f;

  auto issue = [&](int tt) {
    const int b = tt % PIPE;
#pragma unroll
    for (int u = 0; u < K_LD; ++u) {
      const int slot = u * BLK + tid;
      int f2 = tt * TILE_F2 + slot;
      if (f2 >= nf2) f2 = nf2 - 1;               // clamp tail (dup loads, ignored)
      const uint32_t gbyte = (uint32_t)f2 * 8u;
      const uint32_t lb    = lds_base + (uint32_t)((b * TILE_F2 + slot) * 8);
      async_ld_f2(lb, gbyte, base);
    }
  };

  auto consume = [&](int tt) {
    const int b = tt % PIPE;
#pragma unroll
    for (int u = 0; u < K_LD; ++u) {
      const int slot = u * BLK + tid;
      const int f2   = tt * TILE_F2 + slot;
      if (f2 < nf2) {
        const float2 v = lbuf2[b * TILE_F2 + slot];   // ds_load_b64 (own wave's data)
        const int j0 = f2 * 2;
        acc += elem_term(v.x, j0,     target, &s_tgt);
        acc += elem_term(v.y, j0 + 1, target, &s_tgt);
      }
    }
  };

  // ---- prologue: fill the ring ----
  const int npro = ntile < PIPE ? ntile : PIPE;
  for (int p = 0; p < npro; ++p) issue(p);

  // ---- steady state: wait for tile t ((PIPE-1)*K_LD newer ops may remain),
  //      consume it, drain DS reads, refill its slot with tile t+PIPE ----
  int t = 0;
  for (; t + PIPE < ntile; ++t) {
    S_WAIT_ASYNC(12);               // (PIPE-1)*K_LD = 12
    consume(t);
    S_WAIT_DS0();                   // buffer reads done before async rewrite
    issue(t + PIPE);
  }
  // ---- drain ----
  S_WAIT_ASYNC(0);
  for (; t < ntile; ++t) consume(t);

  // ---- odd-column tail (not hit for C = 85742, kept for generality) ----
  if ((ncols & 1) && tid == 0) {
    const int j = ncols - 1;
    acc += elem_term(base[j], j, target, &s_tgt);
  }

  // ---- deterministic block reduction ----
  red[tid] = acc;
  __syncthreads();
#pragma unroll
  for (int s = BLK / 2; s > 0; s >>= 1) {
    if (tid < s) red[tid] += red[tid + s];
    __syncthreads();
  }
  if (tid == 0) {
    // fixed-shift logsumexp: logz = log(sum exp(x-32)) + 32
    row_loss[row] = logf(red[0]) + SCALING - s_tgt;
  }
}

// Deterministic mean over the per-row losses.
__global__ __launch_bounds__(256) void arcface_mean_kernel(
    const float* __restrict__ row_loss, float* __restrict__ out, int nrows)
{
  __shared__ float red[256];
  float a = 0.0f;
  for (int i = threadIdx.x; i < nrows; i += 256) a += row_loss[i];
  red[threadIdx.x] = a;
  __syncthreads();
#pragma unroll
  for (int s = 128; s > 0; s >>= 1) {
    if (threadIdx.x < s) red[threadIdx.x] += red[threadIdx.x + s];
    __syncthreads();
  }
  if (threadIdx.x == 0) out[0] = red[0] / (float)nrows;
}

extern "C" void kernel_launch(void* const* d_in, const int* in_sizes, int n_in,
                              void* d_out, int out_size, void* d_ws, size_t ws_size,
                              hipStream_t stream) {
  const float* cosine  = (const float*)d_in[0];
  const int*   targets = (const int*)d_in[1];
  const int    nrows   = in_sizes[1];
  const int    ncols   = in_sizes[0] / nrows;

  float* row_loss = (float*)d_ws;   // nrows floats of scratch

  arcface_row_kernel<<<nrows, BLK, 0, stream>>>(cosine, targets, row_loss, ncols);
  arcface_mean_kernel<<<1, 256, 0, stream>>>(row_loss, (float*)d_out, nrows);
}